// ResidualBlock_35716948033985
// MI455X (gfx1250) — compile-verified
//
#include <hip/hip_runtime.h>
#include <math.h>

typedef float v2f __attribute__((ext_vector_type(2)));
typedef float v8f __attribute__((ext_vector_type(8)));

#define Hc 128
#define Bc 16
#define Tc 16384
#define Sc 256
#define TT 64      // time-tile per workgroup
#define SX 80      // xs LDS row stride (floats): row+1 bank shift = 16 -> conflict free
#define SG 84      // gs LDS row stride: shifts 20/40/32 for +1/+2/+8 rows -> conflict free

static __device__ __forceinline__ v8f wmma_f32(v2f a, v2f b, v8f c) {
  // D = A(16x4) * B(4x16) + C, all f32, wave32
  return __builtin_amdgcn_wmma_f32_16x16x4_f32(false, a, false, b, (short)0, c,
                                               false, false);
}

static __device__ __forceinline__ v8f vzero8() {
  v8f z;
#pragma unroll
  for (int i = 0; i < 8; ++i) z[i] = 0.0f;
  return z;
}

// branch-free gated activation: tanh(yL) * sigmoid(yR), all native VALU ops
static __device__ __forceinline__ float gate(float yL, float yR) {
  float th = 1.0f - 2.0f * __builtin_amdgcn_rcpf(1.0f + __expf(yL + yL));
  float sg = __builtin_amdgcn_rcpf(1.0f + __expf(-yR));
  return th * sg;
}

__global__ __launch_bounds__(256, 2)
void wavenet_block(const float* __restrict__ input_,   // [H][B][T]
                   const float* __restrict__ cond,     // [2H][B][T]
                   const float* __restrict__ Wd,       // [2H][H][2]
                   const float* __restrict__ bd,       // [2H]
                   const float* __restrict__ Wo,       // [H][H]
                   const float* __restrict__ bo,       // [H]
                   const float* __restrict__ Wsk,      // [S][H]
                   const float* __restrict__ bsk,      // [S]
                   float* __restrict__ out,            // [B][H][T]
                   float* __restrict__ skp)            // [B][S][T]
{
  extern __shared__ float smem[];
  float* xs = smem;            // [128][SX], col c holds x[., t0-2+c], c in [0,66)
  float* gs = smem + Hc * SX;  // [128][SG], gated activations

  const int tid  = threadIdx.x;
  const int w    = tid >> 5;
  const int lane = tid & 31;
  const int lo   = lane & 15;   // N (col) / M (row) index within fragment
  const int hi   = lane >> 4;   // half-wave select

  const int nt = Tc / TT;
  const int bi = blockIdx.x / nt;
  const int t0 = (blockIdx.x % nt) * TT;

  // ---------------- Phase 1: stage x tile (causal left pad 2) ----------------
  {
    const float* xin = input_ + (size_t)bi * Tc;  // + h*(B*T) + t
    for (int i = tid; i < Hc * (TT + 2); i += 256) {
      int r = i / (TT + 2);
      int c = i - r * (TT + 2);
      int t = t0 - 2 + c;
      float v = (t >= 0) ? xin[(size_t)r * (Bc * Tc) + t] : 0.0f;
      xs[r * SX + c] = v;
    }
  }
  __syncthreads();

  // ------------- Phase 2: dilated conv (K=256 GEMM) + gate -> gs -------------
  {
    const int hL = 16 * w;        // tanh half rows (also g rows)
    const int hR = Hc + hL;       // sigmoid half rows
    v8f accL[4], accR[4];
#pragma unroll
    for (int tt = 0; tt < 4; ++tt) { accL[tt] = vzero8(); accR[tt] = vzero8(); }

    // A[o][kappa=2h+c] = Wd[o*256 + kappa] : contiguous pair -> b64 loads
    const float* aLb = Wd + (size_t)(hL + lo) * 256 + 2 * hi;
    const float* aRb = Wd + (size_t)(hR + lo) * 256 + 2 * hi;

    for (int j = 0; j < 64; ++j) {          // kappa block = 4j, h rows 2j, 2j+1
      v2f aL = *(const v2f*)(aLb + 4 * j);
      v2f aR = *(const v2f*)(aRb + 4 * j);
      const float* bp = xs + (2 * j + hi) * SX + lo;
#pragma unroll
      for (int tt = 0; tt < 4; ++tt) {
        v2f bx;                              // v0 = x[t-2], v1 = x[t]
        bx.x = bp[16 * tt];
        bx.y = bp[16 * tt + 2];
        accL[tt] = wmma_f32(aL, bx, accL[tt]);
        accR[tt] = wmma_f32(aR, bx, accR[tt]);
      }
    }

    // bias + conditional + branch-free gating, write g to LDS
    const float* cb = cond + (size_t)bi * Tc;  // + ch*(B*T) + t
#pragma unroll
    for (int r = 0; r < 8; ++r) {
      int gr = r + 8 * hi;
      int oL = hL + gr, oR = hR + gr;
      float bL = bd[oL], bR = bd[oR];
#pragma unroll
      for (int tt = 0; tt < 4; ++tt) {
        int t = t0 + 16 * tt + lo;
        float yL = accL[tt][r] + bL + cb[(size_t)oL * (Bc * Tc) + t];
        float yR = accR[tt][r] + bR + cb[(size_t)oR * (Bc * Tc) + t];
        gs[oL * SG + 16 * tt + lo] = gate(yL, yR);
      }
    }
  }
  __syncthreads();

  // ---- Phase 3: fused skip (256x64) + out (128x64) GEMMs, shared B-frags ----
  {
    v8f acc[3][4];                 // tile-rows: 0,1 = skip, 2 = out
#pragma unroll
    for (int tr = 0; tr < 3; ++tr)
#pragma unroll
      for (int tt = 0; tt < 4; ++tt) acc[tr][tt] = vzero8();

    const int s0 = 32 * w;         // skip rows for this wave
    const int o0 = 16 * w;         // out rows for this wave
    const float* a0 = Wsk + (size_t)(s0 + lo) * Hc + 2 * hi;
    const float* a1 = Wsk + (size_t)(s0 + 16 + lo) * Hc + 2 * hi;
    const float* a2 = Wo  + (size_t)(o0 + lo) * Hc + 2 * hi;

    for (int j = 0; j < 32; ++j) {          // h block = 4j
      v2f A0 = *(const v2f*)(a0 + 4 * j);
      v2f A1 = *(const v2f*)(a1 + 4 * j);
      v2f A2 = *(const v2f*)(a2 + 4 * j);
      const float* gp = gs + (4 * j + 2 * hi) * SG + lo;
#pragma unroll
      for (int tt = 0; tt < 4; ++tt) {
        v2f bx;
        bx.x = gp[16 * tt];        // row h
        bx.y = gp[SG + 16 * tt];   // row h+1
        acc[0][tt] = wmma_f32(A0, bx, acc[0][tt]);
        acc[1][tt] = wmma_f32(A1, bx, acc[1][tt]);
        acc[2][tt] = wmma_f32(A2, bx, acc[2][tt]);
      }
    }

    // ---- store skip = acc[0..1] + bsk ----
    float* sk = skp + (size_t)bi * (Sc * Tc);
#pragma unroll
    for (int tr = 0; tr < 2; ++tr)
#pragma unroll
      for (int r = 0; r < 8; ++r) {
        int s = s0 + 16 * tr + r + 8 * hi;
        float bias = bsk[s];
#pragma unroll
        for (int tt = 0; tt < 4; ++tt) {
          int t = t0 + 16 * tt + lo;
          sk[(size_t)s * Tc + t] = acc[tr][tt][r] + bias;
        }
      }

    // ---- store out = acc[2] + bo + original x (from LDS) ----
    float* op = out + (size_t)bi * (Hc * Tc);
#pragma unroll
    for (int r = 0; r < 8; ++r) {
      int o = o0 + r + 8 * hi;
      float bias = bo[o];
#pragma unroll
      for (int tt = 0; tt < 4; ++tt) {
        int t = t0 + 16 * tt + lo;
        float v = acc[2][tt][r] + bias + xs[o * SX + (16 * tt + lo + 2)];
        op[(size_t)o * Tc + t] = v;
      }
    }
  }
}

extern "C" void kernel_launch(void* const* d_in, const int* in_sizes, int n_in,
                              void* d_out, int out_size, void* d_ws, size_t ws_size,
                              hipStream_t stream) {
  const float* input_ = (const float*)d_in[0];
  const float* cond   = (const float*)d_in[1];
  const float* Wd     = (const float*)d_in[2];
  const float* bd     = (const float*)d_in[3];
  const float* Wo     = (const float*)d_in[4];
  const float* bo     = (const float*)d_in[5];
  const float* Wsk    = (const float*)d_in[6];
  const float* bsk    = (const float*)d_in[7];

  float* out = (float*)d_out;                         // [B][H][T]
  float* skp = out + (size_t)Bc * Hc * Tc;            // [B][S][T] follows

  const int grid = Bc * (Tc / TT);                    // 4096 workgroups
  const size_t smem = (size_t)(Hc * SX + Hc * SG) * sizeof(float);  // 83968 B

  hipLaunchKernelGGL(wavenet_block, dim3(grid), dim3(256), smem, stream,
                     input_, cond, Wd, bd, Wo, bo, Wsk, bsk, out, skp);
}